// BaseSigKernel_60335700574840
// MI455X (gfx1250) — compile-verified
//
#include <hip/hip_runtime.h>

// CDNA5 / gfx1250 signature-kernel Gram matrix.
// xs: (16,128,8) f32, ys: (16,128,8) f32  ->  out: (16,16) f32.
// One workgroup (256 threads = 8 wave32) per (x,y) pair.

typedef __attribute__((ext_vector_type(2))) float v2f;
typedef __attribute__((ext_vector_type(8))) float v8f;

#define PLEN 128   // path length
#define FDIM 8     // feature dim
#define MGRID 254  // refined grid size (M = N = (128-1)*2)
#define NDIAG 507  // number of wavefront steps: d = 2 .. 508

__global__ __launch_bounds__(256)
void sigkernel_pair_kernel(const float* __restrict__ xs,
                           const float* __restrict__ ys,
                           float* __restrict__ out) {
  // LDS: K Gram (128x129 padded) + row norms + 3 rotating diagonals  ~70 KB
  __shared__ float Ks[PLEN][PLEN + 1];
  __shared__ float xx[PLEN];
  __shared__ float yy[PLEN];
  __shared__ float diag[3][256];

  const int b  = blockIdx.x;       // 0..255
  const int ix = b >> 4;
  const int iy = b & 15;
  const float* __restrict__ x = xs + (size_t)ix * PLEN * FDIM;
  const float* __restrict__ y = ys + (size_t)iy * PLEN * FDIM;

  const int tid  = threadIdx.x;
  const int lane = tid & 31;
  const int wave = tid >> 5;       // 0..7

  // ---- row squared norms (threads 0..127 -> xx, 128..255 -> yy) ----
  {
    const float* src = (tid < PLEN) ? x : y;
    int r = (tid < PLEN) ? tid : (tid - PLEN);
    float s = 0.f;
#pragma unroll
    for (int k = 0; k < FDIM; ++k) {
      float v = src[r * FDIM + k];
      s += v * v;
    }
    if (tid < PLEN) xx[r] = s; else yy[r] = s;
  }
  // init PDE diagonals d=0, d=1 to the boundary value 1
  diag[0][tid] = 1.0f;
  diag[1][tid] = 1.0f;
  __syncthreads();

  // ---- RBF Gram via V_WMMA_F32_16X16X4_F32, K=8 as two k-steps ----
  // A 16x4 f32 layout: lane half (lane>>4) selects K pair {0,1}/{2,3};
  //   VGPR0 = K=kb, VGPR1 = K=kb+1; row M = lane&15.  B 4x16 mirrors it.
  const int half = lane >> 4;      // 0 or 1
  const int lr   = lane & 15;      // tile row (A) / tile col (B)
  const int kb   = half * 2;

  const int ti = wave;             // this wave owns output tile row ti
  const float* xr = x + (ti * 16 + lr) * FDIM;
  v2f a0, a1;
  a0.x = xr[kb];     a0.y = xr[kb + 1];       // K = 0..3 slice
  a1.x = xr[4 + kb]; a1.y = xr[4 + kb + 1];   // K = 4..7 slice

#pragma unroll
  for (int tj = 0; tj < 8; ++tj) {
    const float* yr = y + (tj * 16 + lr) * FDIM;
    v2f b0, b1;
    b0.x = yr[kb];     b0.y = yr[kb + 1];
    b1.x = yr[4 + kb]; b1.y = yr[4 + kb + 1];

    v8f acc = {};
    acc = __builtin_amdgcn_wmma_f32_16x16x4_f32(false, a0, false, b0,
                                                (short)0, acc, false, false);
    acc = __builtin_amdgcn_wmma_f32_16x16x4_f32(false, a1, false, b1,
                                                (short)0, acc, false, false);

    // D layout: VGPR v holds (M = v + 8*half, N = lr) of the 16x16 tile.
    const int n = tj * 16 + lr;
    const float yn = yy[n];
#pragma unroll
    for (int v = 0; v < 8; ++v) {
      int m = ti * 16 + v + 8 * half;
      float sq = xx[m] + yn - 2.0f * acc[v];
      Ks[m][n] = __expf(-0.5f * sq);    // sigma = 1
    }
  }
  __syncthreads();

  // ---- Goursat PDE wavefront over the 255x255 grid ----
  // inc at refined cell (I,J) (I,J in 0..253) is
  //   0.25 * (K[p+1][q+1] + K[p][q] - K[p+1][q] - K[p][q+1]), p=I>>1, q=J>>1.
  float* p2 = diag[0];   // diagonal d-2
  float* p1 = diag[1];   // diagonal d-1
  float* cu = diag[2];   // diagonal d (being written)

  const int i = tid;     // row index along the diagonal, active for i<=254
  for (int s = 0; s < NDIAG; ++s) {
    const int d = s + 2;
    if (i <= MGRID) {
      const int j = d - i;
      float val = 1.0f;  // boundary (i==0 || j==0) and outside-grid cells
      if (i >= 1 && j >= 1 && j <= MGRID) {
        const int p = (i - 1) >> 1;
        const int q = (j - 1) >> 1;
        float inc = 0.25f * (Ks[p + 1][q + 1] + Ks[p][q]
                             - Ks[p + 1][q] - Ks[p][q + 1]);
        float k10 = p1[i - 1];
        float k01 = p1[i];
        float k00 = p2[i - 1];
        float i2  = inc * inc * (1.0f / 12.0f);
        val = (k10 + k01) * (1.0f + 0.5f * inc + i2) - k00 * (1.0f - i2);
      }
      cu[i] = val;
    }
    // rotate buffers
    float* t = p2; p2 = p1; p1 = cu; cu = t;
    __syncthreads();
  }

  // final diagonal is in p1 after the last rotation; answer at i = M = 254
  if (tid == 0) out[b] = p1[MGRID];
}

extern "C" void kernel_launch(void* const* d_in, const int* in_sizes, int n_in,
                              void* d_out, int out_size, void* d_ws, size_t ws_size,
                              hipStream_t stream) {
  (void)in_sizes; (void)n_in; (void)d_ws; (void)ws_size; (void)out_size;
  const float* xs = (const float*)d_in[0];
  const float* ys = (const float*)d_in[1];
  float* out = (float*)d_out;
  sigkernel_pair_kernel<<<256, 256, 0, stream>>>(xs, ys, out);
}